// LiftedStructuredEmbedding_23493471109713
// MI455X (gfx1250) — compile-verified
//
#include <hip/hip_runtime.h>
#include <hip/hip_bf16.h>

typedef __attribute__((ext_vector_type(2))) float v2f;
typedef __attribute__((ext_vector_type(8))) float v8f;

#define N_TOT   4096
#define DIMS    128
#define KCLS    4
#define PCLS    1024
#define NPAIR   6144          // PCLS * 6
#define MARGIN_F 1.0f
#define EPS_F    1e-6f

// ---------------------------------------------------------------------------
// Kernel 1: row norms |x_i|^2, and zero the scalar output.
// ---------------------------------------------------------------------------
__global__ __launch_bounds__(256) void lse_norms_kernel(const float* __restrict__ x,
                                                        float* __restrict__ norms,
                                                        float* __restrict__ out) {
  const int i = blockIdx.x * blockDim.x + threadIdx.x;
  if (i == 0) out[0] = 0.0f;
  if (i < N_TOT) {
    const float4* xp = (const float4*)(x + (size_t)i * DIMS);
    float s = 0.0f;
#pragma unroll
    for (int t = 0; t < DIMS / 4; ++t) {
      float4 v = xp[t];
      s += v.x * v.x + v.y * v.y + v.z * v.z + v.w * v.w;
    }
    norms[i] = s;
  }
}

// ---------------------------------------------------------------------------
// Kernel 2: fused fp32 WMMA Gram + distance + masked exp row-sum.
// One block per 16-row tile of D. 8 waves/block; wave w handles column tiles
// w, w+8, ... (32 tiles each). A panel (16x128) held in registers per wave.
// ---------------------------------------------------------------------------
__global__ __launch_bounds__(256) void lse_gemm_rowsum_kernel(const float* __restrict__ x,
                                                              const float* __restrict__ norms,
                                                              float* __restrict__ E,
                                                              float* __restrict__ Dpos) {
  const int lane = threadIdx.x & 31;
  const int wave = threadIdx.x >> 5;        // 0..7
  const int r    = lane & 15;               // row/col slot within tile
  const int hi   = lane >> 4;               // 0: K 0..1 / rows 0..7 ; 1: K 2..3 / rows 8..15
  const int rowBase = blockIdx.x * 16;

  __shared__ float E_lds[16];
  if (threadIdx.x < 16) E_lds[threadIdx.x] = 0.0f;
  __syncthreads();

  // Preload A fragments for this wave's 16x128 row panel.
  // ISA 32-bit A 16x4 layout: lanes 0-15 rows M, VGPR0=K0,VGPR1=K1; lanes 16-31 K2,K3.
  v2f afrag[DIMS / 4];
#pragma unroll
  for (int kk = 0; kk < DIMS / 4; ++kk) {
    afrag[kk] = *(const v2f*)(x + (size_t)(rowBase + r) * DIMS + kk * 4 + hi * 2);
  }
  // Row norms for the 8 rows this lane owns in the C layout (M = v + 8*hi).
  float nrow[8];
#pragma unroll
  for (int v = 0; v < 8; ++v) nrow[v] = norms[rowBase + v + 8 * hi];

  float rowAcc[8];
#pragma unroll
  for (int v = 0; v < 8; ++v) rowAcc[v] = 0.0f;

  for (int ct = wave; ct < N_TOT / 16; ct += 8) {
    const int colBase = ct * 16;
    v8f c = {};
#pragma unroll
    for (int kk = 0; kk < DIMS / 4; ++kk) {
      // B = X^T tile: lane's column = x row (colBase + r); same K packing as A.
      v2f b = *(const v2f*)(x + (size_t)(colBase + r) * DIMS + kk * 4 + hi * 2);
      c = __builtin_amdgcn_wmma_f32_16x16x4_f32(false, afrag[kk], false, b,
                                                (short)0, c, false, false);
    }

    const int   j     = colBase + r;      // this lane's global column
    const float ncol  = norms[j];
    const int   cls_j = j >> 2;

#pragma unroll
    for (int v = 0; v < 8; ++v) {
      const int   i     = rowBase + v + 8 * hi;   // C layout: VGPR v holds rows v / v+8
      const float g     = c[v];
      const float d2    = nrow[v] + ncol - 2.0f * g;
      const float dist  = sqrtf(fmaxf(d2, 0.0f) + EPS_F);
      const int   cls_i = i >> 2;
      if (cls_i != cls_j) {
        rowAcc[v] += __expf(MARGIN_F - dist);     // bounded by e; no max-shift needed
      } else {
        // same class => diagonal tile; capture positive-pair distance (i<j order)
        const int a = i & 3, b2 = j & 3;
        if (a < b2) {
          const int m = a * KCLS - (a * (a + 1)) / 2 + (b2 - a - 1); // triu(4,1) index
          Dpos[cls_i * 6 + m] = dist;
        }
      }
    }
  }

  // Reduce each row across its 16-lane half (xor 8,4,2,1 stays within halves).
#pragma unroll
  for (int v = 0; v < 8; ++v) {
    float s = rowAcc[v];
    s += __shfl_xor(s, 8);
    s += __shfl_xor(s, 4);
    s += __shfl_xor(s, 2);
    s += __shfl_xor(s, 1);
    if ((lane & 15) == 0) atomicAdd(&E_lds[v + 8 * hi], s);
  }
  __syncthreads();
  if (threadIdx.x < 16) E[rowBase + threadIdx.x] = E_lds[threadIdx.x];
}

// ---------------------------------------------------------------------------
// Kernel 3: per-pair loss + global reduction.
// ---------------------------------------------------------------------------
__global__ __launch_bounds__(256) void lse_pair_kernel(const float* __restrict__ E,
                                                       const float* __restrict__ Dpos,
                                                       float* __restrict__ out) {
  const int p = blockIdx.x * blockDim.x + threadIdx.x;  // 0..6143 exactly
  float val = 0.0f;
  if (p < NPAIR) {
    const int c = p / 6;
    const int m = p % 6;
    const int a  = (m < 3) ? 0 : ((m < 5) ? 1 : 2);
    const int b  = (m < 3) ? (m + 1) : ((m < 5) ? (m - 1) : 3);
    const float s = E[c * KCLS + a] + E[c * KCLS + b];
    const float J = logf(s) + Dpos[p];
    const float rl = fmaxf(J, 0.0f);
    val = rl * rl;
  }
  // wave32 reduction
  val += __shfl_xor(val, 16);
  val += __shfl_xor(val, 8);
  val += __shfl_xor(val, 4);
  val += __shfl_xor(val, 2);
  val += __shfl_xor(val, 1);
  __shared__ float wsum[8];
  const int lane = threadIdx.x & 31;
  const int wave = threadIdx.x >> 5;
  if (lane == 0) wsum[wave] = val;
  __syncthreads();
  if (threadIdx.x < 8) {
    float t = wsum[threadIdx.x];
    t += __shfl_xor(t, 4);
    t += __shfl_xor(t, 2);
    t += __shfl_xor(t, 1);
    if (threadIdx.x == 0) {
      atomicAdd(out, t * (1.0f / (2.0f * (float)NPAIR)));  // mean(...)/2
    }
  }
}

// ---------------------------------------------------------------------------
extern "C" void kernel_launch(void* const* d_in, const int* in_sizes, int n_in,
                              void* d_out, int out_size, void* d_ws, size_t ws_size,
                              hipStream_t stream) {
  (void)in_sizes; (void)n_in; (void)out_size; (void)ws_size;
  const float* x   = (const float*)d_in[0];
  float*       out = (float*)d_out;

  float* norms = (float*)d_ws;            // 4096 f32
  float* E     = norms + N_TOT;           // 4096 f32
  float* Dpos  = E + N_TOT;               // 6144 f32  (total 56 KB of ws)

  lse_norms_kernel<<<N_TOT / 256, 256, 0, stream>>>(x, norms, out);
  lse_gemm_rowsum_kernel<<<N_TOT / 16, 256, 0, stream>>>(x, norms, E, Dpos);
  lse_pair_kernel<<<NPAIR / 256, 256, 0, stream>>>(E, Dpos, out);
}